// Listener0Model_39152921870344
// MI455X (gfx1250) — compile-verified
//
#include <hip/hip_runtime.h>

#define H        256
#define BSZ      4096
#define PP       20
#define AALT     9
#define LL       50
#define TT       10      // 1 + A
#define APAD     260     // LDS row pitch (floats): stride 1040B -> conflict-free K reads

typedef __attribute__((ext_vector_type(2))) float v2f;
typedef __attribute__((ext_vector_type(8))) float v8f;

// -------- Kernel 1: s5buf[b, :] = (b_str + sum_l W_str[word[b,l]]) @ W5 + b5 --------
__global__ __launch_bounds__(256) void k_s5(
    const int*   __restrict__ word_idx,
    const float* __restrict__ W_str,  const float* __restrict__ b_str,
    const float* __restrict__ W5,     const float* __restrict__ b5,
    float*       __restrict__ s5buf)
{
  __shared__ float At[16 * APAD];
  __shared__ int   sw[16 * LL];
  const int tid = threadIdx.x;
  const int m0  = blockIdx.x * 16;

  for (int i = tid; i < 16 * LL; i += 256) {
    int r = i / LL, j = i % LL;
    sw[i] = word_idx[(m0 + r) * LL + j];
  }
  __syncthreads();

  // Stage 16 string-encoder rows into LDS (bag-of-words: duplicates DO count)
  for (int r = 0; r < 16; ++r) {
    float a = b_str[tid];
    #pragma unroll 5
    for (int j = 0; j < LL; ++j)
      a += W_str[sw[r * LL + j] * H + tid];
    At[r * APAD + tid] = a;
  }
  __syncthreads();

  // WMMA f32 GEMM: rows m0..m0+15, wave w -> cols [32w, 32w+32)
  const int lane  = tid & 31;
  const int wv    = tid >> 5;
  const int lnm   = lane & 15;
  const int khalf = lane >> 4;
  const int n0    = wv * 32;

  v8f c0, c1;
  const float bias0 = b5[n0 + lnm];
  const float bias1 = b5[n0 + 16 + lnm];
  #pragma unroll
  for (int r = 0; r < 8; ++r) { c0[r] = bias0; c1[r] = bias1; }

  #pragma unroll 2
  for (int kk = 0; kk < H / 4; ++kk) {
    const int k0 = kk * 4;
    const int kb = k0 + 2 * khalf;
    v2f a, b0, b1;
    a.x  = At[lnm * APAD + kb];
    a.y  = At[lnm * APAD + kb + 1];
    b0.x = W5[kb * H + n0 + lnm];
    b0.y = W5[(kb + 1) * H + n0 + lnm];
    b1.x = W5[kb * H + n0 + 16 + lnm];
    b1.y = W5[(kb + 1) * H + n0 + 16 + lnm];
    c0 = __builtin_amdgcn_wmma_f32_16x16x4_f32(false, a, false, b0, (short)0, c0, false, false);
    c1 = __builtin_amdgcn_wmma_f32_16x16x4_f32(false, a, false, b1, (short)0, c1, false, false);
  }

  #pragma unroll
  for (int r = 0; r < 8; ++r) {
    const int row = r + 8 * khalf;
    s5buf[(m0 + row) * H + n0 + lnm]      = c0[r];
    s5buf[(m0 + row) * H + n0 + 16 + lnm] = c1[r];
  }
}

// -------- Kernel 2: fused targets-build + GEMM(W4) + relu(+s5) + dot(W3) --------
__global__ __launch_bounds__(256) void k_score(
    const int*   __restrict__ prop_idx,
    const int*   __restrict__ alt_prop_idx,
    const float* __restrict__ W_scene, const float* __restrict__ b_scene,
    const float* __restrict__ W4,      const float* __restrict__ b4,
    const float* __restrict__ W3,      const float* __restrict__ b3,
    const float* __restrict__ s5buf,
    float*       __restrict__ out)
{
  __shared__ float At[16 * APAD];
  __shared__ float Post[16 * APAD];
  __shared__ int   sidx[16 * PP];
  __shared__ int   skeep[16 * PP];
  __shared__ float red[16 * 16];
  const int tid = threadIdx.x;
  const int m0  = blockIdx.x * 16;   // global row = b*T + t

  // Gather prop indices for 16 (b,t) rows
  for (int i = tid; i < 16 * PP; i += 256) {
    int r = i / PP, j = i % PP;
    int grow = m0 + r;
    int bb = grow / TT, tt = grow % TT;
    sidx[i] = (tt == 0) ? prop_idx[bb * PP + j]
                        : alt_prop_idx[((tt - 1) * BSZ + bb) * PP + j];
  }
  __syncthreads();
  // Dedupe (one-hot .set semantics: repeated props count once)
  for (int i = tid; i < 16 * PP; i += 256) {
    int r = i / PP, j = i % PP;
    int v = sidx[i], keep = 1;
    for (int q = 0; q < j; ++q)
      if (sidx[r * PP + q] == v) { keep = 0; break; }
    skeep[i] = keep;
  }
  __syncthreads();
  // Stage A rows: scene encodings (column tid for all 16 rows; coalesced W_scene reads)
  for (int r = 0; r < 16; ++r) {
    float a = b_scene[tid];
    for (int j = 0; j < PP; ++j)
      if (skeep[r * PP + j]) a += W_scene[sidx[r * PP + j] * H + tid];
    At[r * APAD + tid] = a;
  }
  __syncthreads();

  const int lane  = tid & 31;
  const int wv    = tid >> 5;
  const int lnm   = lane & 15;
  const int khalf = lane >> 4;
  const int n0    = wv * 32;

  v8f c0, c1;
  const float bias0 = b4[n0 + lnm];
  const float bias1 = b4[n0 + 16 + lnm];
  #pragma unroll
  for (int r = 0; r < 8; ++r) { c0[r] = bias0; c1[r] = bias1; }

  #pragma unroll 2
  for (int kk = 0; kk < H / 4; ++kk) {
    const int k0 = kk * 4;
    const int kb = k0 + 2 * khalf;
    v2f a, b0, b1;
    a.x  = At[lnm * APAD + kb];
    a.y  = At[lnm * APAD + kb + 1];
    b0.x = W4[kb * H + n0 + lnm];
    b0.y = W4[(kb + 1) * H + n0 + lnm];
    b1.x = W4[kb * H + n0 + 16 + lnm];
    b1.y = W4[(kb + 1) * H + n0 + 16 + lnm];
    c0 = __builtin_amdgcn_wmma_f32_16x16x4_f32(false, a, false, b0, (short)0, c0, false, false);
    c1 = __builtin_amdgcn_wmma_f32_16x16x4_f32(false, a, false, b1, (short)0, c1, false, false);
  }

  // Epilogue: + s5[b] broadcast over t, ReLU, stash in LDS
  #pragma unroll
  for (int r = 0; r < 8; ++r) {
    const int row  = r + 8 * khalf;
    const int grow = m0 + row;
    const int bb   = grow / TT;
    float v0 = c0[r] + s5buf[bb * H + n0 + lnm];
    float v1 = c1[r] + s5buf[bb * H + n0 + 16 + lnm];
    Post[row * APAD + n0 + lnm]      = v0 > 0.f ? v0 : 0.f;
    Post[row * APAD + n0 + 16 + lnm] = v1 > 0.f ? v1 : 0.f;
  }
  __syncthreads();

  // Block-local dot with W3: 16 threads per row, 16 cols each
  {
    const int row = tid >> 4, seg = tid & 15;
    float p = 0.f;
    #pragma unroll
    for (int j = 0; j < 16; ++j) {
      const int h = seg * 16 + j;
      p += Post[row * APAD + h] * W3[h];
    }
    red[row * 16 + seg] = p;
  }
  __syncthreads();
  if (tid < 16) {
    float s = b3[0];
    #pragma unroll
    for (int seg = 0; seg < 16; ++seg) s += red[tid * 16 + seg];
    out[m0 + tid] = s;   // out is [B,T] row-major == global row index
  }
}

extern "C" void kernel_launch(void* const* d_in, const int* in_sizes, int n_in,
                              void* d_out, int out_size, void* d_ws, size_t ws_size,
                              hipStream_t stream) {
  const int*   prop_idx     = (const int*)d_in[0];
  const int*   alt_prop_idx = (const int*)d_in[1];
  const int*   word_idx     = (const int*)d_in[2];
  const float* W_scene      = (const float*)d_in[3];
  const float* b_scene      = (const float*)d_in[4];
  const float* W_str        = (const float*)d_in[5];
  const float* b_str        = (const float*)d_in[6];
  const float* W4           = (const float*)d_in[7];
  const float* b4           = (const float*)d_in[8];
  const float* W5           = (const float*)d_in[9];
  const float* b5           = (const float*)d_in[10];
  const float* W3           = (const float*)d_in[11];
  const float* b3           = (const float*)d_in[12];
  float* s5buf = (float*)d_ws;                  // B*H floats = 4 MB
  float* out   = (float*)d_out;                 // [B, T]

  hipLaunchKernelGGL(k_s5, dim3(BSZ / 16), dim3(256), 0, stream,
                     word_idx, W_str, b_str, W5, b5, s5buf);
  hipLaunchKernelGGL(k_score, dim3((BSZ * TT) / 16), dim3(256), 0, stream,
                     prop_idx, alt_prop_idx, W_scene, b_scene, W4, b4, W3, b3,
                     s5buf, out);
}